// SingleHeadAttention_71554155151838
// MI455X (gfx1250) — compile-verified
//
#include <hip/hip_runtime.h>

// ---------------------------------------------------------------------------
// Single-head causal attention for MI455X (gfx1250), exact-fp32 WMMA path.
//   tokens [8192 x 768] fp32 ; Wk/Wq/Wv [768 x 64] fp32 ; out [8192 x 64] fp32
// Stage 1: K/Q/V projections via V_WMMA_F32_16X16X4_F32 (one wave / 16x16 tile)
// Stage 2: flash attention, one wave per 16-row query block, blocks paired
//          (g, 511-g) so every wave does an identical 513 key-tiles of work.
// ---------------------------------------------------------------------------

typedef __attribute__((ext_vector_type(2))) float v2f;
typedef __attribute__((ext_vector_type(8))) float v8f;

#define TT 8192
#define CC 768
#define HH 64
#define NQB (TT / 16) /* 512 query blocks */
#define SCALE 27.712812921102035f /* C ** 0.5 (reference multiplies, not divides) */
#define NEG_BIG (-1.0e30f)

static __device__ __forceinline__ v8f wmma4(v2f a, v2f b, v8f c) {
  // D = A(16x4 f32) * B(4x16 f32) + C(16x16 f32), full fp32 precision.
  return __builtin_amdgcn_wmma_f32_16x16x4_f32(false, a, false, b, (short)0, c,
                                               false, false);
}

// Butterfly reductions across the 16-lane half-wave that holds one matrix row.
static __device__ __forceinline__ float rmax16(float x) {
#pragma unroll
  for (int off = 8; off >= 1; off >>= 1) x = fmaxf(x, __shfl_xor(x, off, 32));
  return x;
}
static __device__ __forceinline__ float rsum16(float x) {
#pragma unroll
  for (int off = 8; off >= 1; off >>= 1) x += __shfl_xor(x, off, 32);
  return x;
}

// ---------------------------------------------------------------------------
// Stage 1: KQV = tokens @ [Wk | Wq | Wv].  One wave per 16x16 output tile.
// Grid: 512 row-tiles * 3 matrices * 4 col-tiles = 6144 waves = 1536 blocks.
// ---------------------------------------------------------------------------
__global__ void __launch_bounds__(128) proj_kernel(
    const float* __restrict__ tok, const float* __restrict__ Wk,
    const float* __restrict__ Wq, const float* __restrict__ Wv,
    float* __restrict__ kqv) {
  const int wid = threadIdx.x >> 5;
  const int lane = threadIdx.x & 31;
  const int g = blockIdx.x * 4 + wid; // 0..6143
  const int rowtile = g / 12;
  const int rem = g % 12;
  const int mat = rem >> 2;    // 0=K, 1=Q, 2=V
  const int coltile = rem & 3; // 4 tiles of 16 cover H=64

  const float* W = (mat == 0) ? Wk : ((mat == 1) ? Wq : Wv);
  float* out = kqv + (size_t)mat * TT * HH;

  const int n = lane & 15;     // A-row m / B-col n (same lane mapping)
  const int half = lane >> 4;  // half-wave select
  const int koff = half * 2;   // K-pair offset within a 4-wide K step

  const float* arow = tok + (size_t)(rowtile * 16 + n) * CC + koff;
  const float* bcol = W + (size_t)koff * HH + coltile * 16 + n;

  v8f acc = {0.f, 0.f, 0.f, 0.f, 0.f, 0.f, 0.f, 0.f};
#pragma unroll 8
  for (int k = 0; k < CC; k += 4) {
    v2f a = *(const v2f*)(arow + k); // contiguous K pair, 8B aligned
    v2f b;
    b.x = bcol[(size_t)k * HH];
    b.y = bcol[(size_t)(k + 1) * HH];
    acc = wmma4(a, b, acc);
  }

#pragma unroll
  for (int r = 0; r < 8; ++r)
    out[(size_t)(rowtile * 16 + r + 8 * half) * HH + coltile * 16 + n] = acc[r];
}

// ---------------------------------------------------------------------------
// Stage 2: flash attention. 64 blocks x 128 threads = 256 waves; wave g does
// query blocks g and 511-g (513 key-tiles each -> perfectly balanced).
// ---------------------------------------------------------------------------
__global__ void __launch_bounds__(128) attn_kernel(
    const float* __restrict__ K, const float* __restrict__ Q,
    const float* __restrict__ V, float* __restrict__ out) {
  __shared__ float plds_all[4 * 16 * 17]; // per-wave 16x16 P slab, padded

  const int wid = threadIdx.x >> 5;
  const int lane = threadIdx.x & 31;
  const int g = blockIdx.x * 4 + wid; // 0..255
  const int n = lane & 15;
  const int half = lane >> 4;
  const int koff = half * 2;
  float* pl = plds_all + wid * (16 * 17);

  for (int pass = 0; pass < 2; ++pass) {
    const int qb = pass ? (NQB - 1 - g) : g;

    // Preload the wave's 16x64 Q tile as 16 A-fragments (kept in VGPRs).
    v2f qf[16];
    const float* qrow = Q + (size_t)(qb * 16 + n) * HH + koff;
#pragma unroll
    for (int s = 0; s < 16; ++s) qf[s] = *(const v2f*)(qrow + s * 4);

    v8f o0 = {0.f, 0.f, 0.f, 0.f, 0.f, 0.f, 0.f, 0.f};
    v8f o1 = o0, o2 = o0, o3 = o0;
    float mrow[8], lrow[8];
#pragma unroll
    for (int r = 0; r < 8; ++r) {
      mrow[r] = NEG_BIG;
      lrow[r] = 0.f;
    }

    for (int j = 0; j <= qb; ++j) {
      // ---- S = (Q @ K_j^T) * sqrt(C) -----------------------------------
      const float* krow = K + (size_t)(j * 16 + n) * HH + koff; // B = K^T
      v8f s = {0.f, 0.f, 0.f, 0.f, 0.f, 0.f, 0.f, 0.f};
#pragma unroll
      for (int kk = 0; kk < 16; ++kk) {
        v2f b = *(const v2f*)(krow + kk * 4);
        s = wmma4(qf[kk], b, s);
      }
#pragma unroll
      for (int r = 0; r < 8; ++r) s[r] *= SCALE;

      // ---- causal mask on the diagonal tile ----------------------------
      if (j == qb) {
#pragma unroll
        for (int r = 0; r < 8; ++r)
          if (n > r + 8 * half) s[r] = NEG_BIG;
      }

      // ---- online softmax update ---------------------------------------
      float alpha[8];
      v8f p;
#pragma unroll
      for (int r = 0; r < 8; ++r) {
        float mn = fmaxf(mrow[r], rmax16(s[r]));
        alpha[r] = __expf(mrow[r] - mn);
        mrow[r] = mn;
        p[r] = __expf(s[r] - mn);
      }
#pragma unroll
      for (int r = 0; r < 8; ++r) {
        lrow[r] = lrow[r] * alpha[r] + rsum16(p[r]);
        o0[r] *= alpha[r];
        o1[r] *= alpha[r];
        o2[r] *= alpha[r];
        o3[r] *= alpha[r];
      }

      // ---- reshape P (C/D layout) -> A-fragments via per-wave LDS ------
#pragma unroll
      for (int r = 0; r < 8; ++r) pl[(r + 8 * half) * 17 + n] = p[r];
      v2f pa[4];
#pragma unroll
      for (int s2 = 0; s2 < 4; ++s2) {
        pa[s2].x = pl[n * 17 + s2 * 4 + koff];
        pa[s2].y = pl[n * 17 + s2 * 4 + koff + 1];
      }

      // ---- O += P @ V_j  (four 16x16 N-tiles, K=16 in 4 steps) ---------
      const float* vb = V + (size_t)(j * 16) * HH;
#pragma unroll
      for (int s2 = 0; s2 < 4; ++s2) {
        const int kd = s2 * 4 + koff;
        v2f b0, b1, b2, b3;
        b0.x = vb[kd * HH + 0 * 16 + n];
        b0.y = vb[(kd + 1) * HH + 0 * 16 + n];
        b1.x = vb[kd * HH + 1 * 16 + n];
        b1.y = vb[(kd + 1) * HH + 1 * 16 + n];
        b2.x = vb[kd * HH + 2 * 16 + n];
        b2.y = vb[(kd + 1) * HH + 2 * 16 + n];
        b3.x = vb[kd * HH + 3 * 16 + n];
        b3.y = vb[(kd + 1) * HH + 3 * 16 + n];
        o0 = wmma4(pa[s2], b0, o0);
        o1 = wmma4(pa[s2], b1, o1);
        o2 = wmma4(pa[s2], b2, o2);
        o3 = wmma4(pa[s2], b3, o3);
      }
    }

    // ---- epilogue: out = O / l ----------------------------------------
#pragma unroll
    for (int r = 0; r < 8; ++r) {
      const float inv = 1.0f / lrow[r];
      const size_t row = (size_t)(qb * 16 + r + 8 * half) * HH;
      out[row + 0 * 16 + n] = o0[r] * inv;
      out[row + 1 * 16 + n] = o1[r] * inv;
      out[row + 2 * 16 + n] = o2[r] * inv;
      out[row + 3 * 16 + n] = o3[r] * inv;
    }
  }
}

// ---------------------------------------------------------------------------
extern "C" void kernel_launch(void* const* d_in, const int* in_sizes, int n_in,
                              void* d_out, int out_size, void* d_ws,
                              size_t ws_size, hipStream_t stream) {
  const float* tokens = (const float*)d_in[0];
  const float* Wk = (const float*)d_in[1];
  const float* Wq = (const float*)d_in[2];
  const float* Wv = (const float*)d_in[3];
  float* ws = (float*)d_ws; // [K | Q | V], each TT*HH fp32 (6 MB total)

  float* Kp = ws;
  float* Qp = ws + (size_t)TT * HH;
  float* Vp = ws + (size_t)2 * TT * HH;

  // 512 row-tiles * 12 (mat,coltile) = 6144 waves -> 1536 blocks of 4 waves.
  proj_kernel<<<1536, 128, 0, stream>>>(tokens, Wk, Wq, Wv, ws);

  // 256 balanced flash-attention waves -> 64 blocks of 4 waves.
  attn_kernel<<<64, 128, 0, stream>>>(Kp, Qp, Vp, (float*)d_out);
}